// FEMGNN_78486232367508
// MI455X (gfx1250) — compile-verified
//
#include <hip/hip_runtime.h>
#include <hip/hip_bf16.h>

typedef __attribute__((ext_vector_type(2))) float v2f;
typedef __attribute__((ext_vector_type(8))) float v8f;

#define HID 128

// ---------------- degree / norm precompute ----------------
__global__ void deg_init_kernel(float* deg, int n) {
    int i = blockIdx.x * 256 + threadIdx.x;
    if (i < n) deg[i] = 1.0f;            // self loop contributes 1
}

__global__ void deg_acc_kernel(const int* dst, float* deg, int e) {
    int i = blockIdx.x * 256 + threadIdx.x;
    if (i < e) atomicAdd(&deg[dst[i]], 1.0f);
}

__global__ void dinv_kernel(float* deg, int n) {
    int i = blockIdx.x * 256 + threadIdx.x;
    if (i < n) deg[i] = rsqrtf(deg[i]);  // deg>0 always (self loop)
}

__global__ void norm_kernel(const int* src, const int* dst, const float* dinv,
                            float* norm, int e) {
    int i = blockIdx.x * 256 + threadIdx.x;
    if (i < e) norm[i] = dinv[src[i]] * dinv[dst[i]];
}

// ---------------- encoder: h = relu(x @ enc_w + enc_b), K=9 ----------------
__global__ void encoder_kernel(const float* __restrict__ x,
                               const float* __restrict__ w,
                               const float* __restrict__ b,
                               float* __restrict__ h, int n) {
    int gid = blockIdx.x * 256 + threadIdx.x;
    if (gid >= n * HID) return;
    int i = gid >> 7;
    int c = gid & (HID - 1);
    float s = b[c];
    #pragma unroll
    for (int k = 0; k < 9; ++k) s += x[i * 9 + k] * w[k * HID + c];
    h[gid] = fmaxf(s, 0.0f);
}

// ---------------- WMMA fp32 matmul: T = H(MxK=128) @ W(128x128) ----------------
__global__ __launch_bounds__(256) void gcn_matmul_wmma(
        const float* __restrict__ H, const float* __restrict__ W,
        float* __restrict__ T, int M) {
    // W permuted into B-fragment order: [ktile(32)][ntile(8)][lane(32)] x float2
    __shared__ float wfrag[32 * 8 * 32 * 2];   // 64 KB
    const int tid = threadIdx.x;

    for (int f = tid; f < 32 * 8 * 32; f += 256) {
        int lane = f & 31;
        int nt   = (f >> 5) & 7;
        int kt   = f >> 8;
        int k0   = kt * 4 + ((lane >> 4) << 1);   // K=0/1 (lanes 0-15), K=2/3 (lanes 16-31)
        int col  = nt * 16 + (lane & 15);
        wfrag[f * 2 + 0] = W[k0 * HID + col];
        wfrag[f * 2 + 1] = W[(k0 + 1) * HID + col];
    }
    __syncthreads();

    const int wave = tid >> 5;
    const int lane = tid & 31;
    const int row0 = (blockIdx.x * 8 + wave) * 16;     // 16-row strip per wave
    const int koff = (lane >> 4) << 1;                 // A-fragment K offset per half-wave
    int rowA = row0 + (lane & 15);
    if (rowA >= M) rowA = M - 1;                       // clamp loads, keep EXEC all-1s

    const float* aptr = H + (size_t)rowA * HID + koff;
    __builtin_prefetch(aptr, 0, 0);                    // global_prefetch_b8
    __builtin_prefetch(aptr + 64, 0, 0);

    v8f acc[8];
    #pragma unroll
    for (int t = 0; t < 8; ++t) acc[t] = (v8f)0.0f;

    for (int kt = 0; kt < 32; ++kt) {
        v2f a = *(const v2f*)(aptr + kt * 4);
        const float* bbase = &wfrag[(kt * 8 * 32 + lane) * 2];
        #pragma unroll
        for (int nt = 0; nt < 8; ++nt) {
            v2f b = *(const v2f*)(bbase + nt * 32 * 2);
            acc[nt] = __builtin_amdgcn_wmma_f32_16x16x4_f32(
                false, a, false, b, (short)0, acc[nt], false, false);
        }
    }

    // C/D layout: VGPR j -> (M=j, N=lane) lanes 0-15 ; (M=j+8, N=lane-16) lanes 16-31
    #pragma unroll
    for (int nt = 0; nt < 8; ++nt) {
        int col = nt * 16 + (lane & 15);
        #pragma unroll
        for (int j = 0; j < 8; ++j) {
            int r = row0 + j + ((lane >> 4) << 3);
            if (r < M) T[(size_t)r * HID + col] = acc[nt][j];
        }
    }
}

// ---------------- self-loop init: agg = t * dinv^2 ----------------
__global__ void self_init_kernel(const float* __restrict__ t,
                                 const float* __restrict__ dinv,
                                 float* __restrict__ agg, int n) {
    int gid = blockIdx.x * 256 + threadIdx.x;
    if (gid >= n * HID) return;
    float di = dinv[gid >> 7];
    agg[gid] = t[gid] * di * di;
}

// ---------------- edge scatter: agg[dst] += t[src] * norm ----------------
__global__ void edge_scatter_kernel(const int* __restrict__ src,
                                    const int* __restrict__ dst,
                                    const float* __restrict__ norm,
                                    const float* __restrict__ t,
                                    float* __restrict__ agg, int e) {
    long long gid = (long long)blockIdx.x * 256 + threadIdx.x;
    long long ed = gid >> 7;
    int c = (int)(gid & (HID - 1));
    if (ed >= e) return;
    int s = src[ed];
    int d = dst[ed];
    atomicAdd(&agg[(size_t)d * HID + c], t[(size_t)s * HID + c] * norm[ed]);
}

// ---------------- bias + relu in place ----------------
__global__ void bias_relu_kernel(float* __restrict__ h,
                                 const float* __restrict__ b, int n) {
    int gid = blockIdx.x * 256 + threadIdx.x;
    if (gid >= n * HID) return;
    h[gid] = fmaxf(h[gid] + b[gid & (HID - 1)], 0.0f);
}

// ---------------- decoder: out = (h @ dec_w + dec_b) * (1 - x[:,3]) ----------------
__global__ void decoder_kernel(const float* __restrict__ h,
                               const float* __restrict__ x,
                               const float* __restrict__ dw,
                               const float* __restrict__ db,
                               float* __restrict__ out, int n) {
    int wid = (blockIdx.x * 256 + threadIdx.x) >> 5;   // one wave32 per node
    int lane = threadIdx.x & 31;
    if (wid >= n) return;
    float s0 = 0.f, s1 = 0.f, s2 = 0.f;
    for (int c = lane; c < HID; c += 32) {
        float hv = h[(size_t)wid * HID + c];
        s0 += hv * dw[c * 3 + 0];
        s1 += hv * dw[c * 3 + 1];
        s2 += hv * dw[c * 3 + 2];
    }
    #pragma unroll
    for (int off = 16; off > 0; off >>= 1) {
        s0 += __shfl_down(s0, off, 32);
        s1 += __shfl_down(s1, off, 32);
        s2 += __shfl_down(s2, off, 32);
    }
    if (lane == 0) {
        float mask = 1.0f - x[(size_t)wid * 9 + 3];
        out[wid * 3 + 0] = (s0 + db[0]) * mask;
        out[wid * 3 + 1] = (s1 + db[1]) * mask;
        out[wid * 3 + 2] = (s2 + db[2]) * mask;
    }
}

extern "C" void kernel_launch(void* const* d_in, const int* in_sizes, int n_in,
                              void* d_out, int out_size, void* d_ws, size_t ws_size,
                              hipStream_t stream) {
    const float* x     = (const float*)d_in[0];
    const int*   edges = (const int*)d_in[1];     // (2, E): src rows then dst rows
    const float* enc_w = (const float*)d_in[2];
    const float* enc_b = (const float*)d_in[3];
    const float* w1    = (const float*)d_in[4];
    const float* b1    = (const float*)d_in[5];
    const float* w2    = (const float*)d_in[6];
    const float* b2    = (const float*)d_in[7];
    const float* w3    = (const float*)d_in[8];
    const float* b3    = (const float*)d_in[9];
    const float* dec_w = (const float*)d_in[10];
    const float* dec_b = (const float*)d_in[11];
    float* out = (float*)d_out;

    const int n = in_sizes[0] / 9;        // 100000
    const int e = in_sizes[1] / 2;        // 1600000
    const int* src = edges;
    const int* dst = edges + e;

    float* ws   = (float*)d_ws;
    float* dinv = ws;                                  // n
    float* norm = ws + n;                              // e
    float* bufA = ws + n + e;                          // n*128
    float* bufB = bufA + (size_t)n * HID;              // n*128

    const int T256 = 256;
    int nb_n   = (n + 255) / 256;
    int nb_e   = (e + 255) / 256;
    int nb_nh  = (int)(((long long)n * HID + 255) / 256);
    long long eh = (long long)e * HID;
    int nb_eh  = (int)((eh + 255) / 256);
    int nb_mm  = (n + 127) / 128;                      // 8 waves x 16 rows per block
    int nb_dec = (int)(((long long)n * 32 + 255) / 256);

    // degrees and edge norms
    deg_init_kernel<<<nb_n, T256, 0, stream>>>(dinv, n);
    deg_acc_kernel<<<nb_e, T256, 0, stream>>>(dst, dinv, e);
    dinv_kernel<<<nb_n, T256, 0, stream>>>(dinv, n);
    norm_kernel<<<nb_e, T256, 0, stream>>>(src, dst, dinv, norm, e);

    // encoder
    encoder_kernel<<<nb_nh, T256, 0, stream>>>(x, enc_w, enc_b, bufA, n);

    // 3 GCN layers: A -> (matmul) B -> (scatter) A
    const float* Ws[3] = {w1, w2, w3};
    const float* Bs[3] = {b1, b2, b3};
    for (int l = 0; l < 3; ++l) {
        gcn_matmul_wmma<<<nb_mm, T256, 0, stream>>>(bufA, Ws[l], bufB, n);
        self_init_kernel<<<nb_nh, T256, 0, stream>>>(bufB, dinv, bufA, n);
        edge_scatter_kernel<<<nb_eh, T256, 0, stream>>>(src, dst, norm, bufB, bufA, e);
        bias_relu_kernel<<<nb_nh, T256, 0, stream>>>(bufA, Bs[l], n);
    }

    // decoder + boundary mask
    decoder_kernel<<<nb_dec, T256, 0, stream>>>(bufA, x, dec_w, dec_b, out, n);
}